// KendallLayer_11733850653132
// MI455X (gfx1250) — compile-verified
//
#include <hip/hip_runtime.h>
#include <hip/hip_bf16.h>

// Problem constants (fixed by setup_inputs)
#define T_DIM   4
#define WQ      375
#define WS      25
#define C_DIM   64
#define HW      25
#define WAY     5
#define SHOT    5
#define NQ      (T_DIM * WQ)          // 1500
#define NS      (T_DIM * WAY)         // 20
#define NPAIR   2016                  // C(64,2)
#define LOG2E   1.4426950408889634f

typedef __attribute__((ext_vector_type(2))) float v2f;
typedef __attribute__((ext_vector_type(8))) float v8f;

// -------------------------------------------------------------------------
// Mean over 25 contiguous elements per row, via V_WMMA_F32_16X16X4_F32.
// A = ones(16x4)  =>  D[m][n] = sum_k B[k][n]  (independent of A layout and
// of K ordering within B's slots). C/D layout: N = lane%16, so lane L holds
// the running sum for row (rowbase + L%16) in every D VGPR.
// 6 WMMAs cover K=0..23 (chunks of 4); element 24 added with VALU.
// Launch with exact grids: nrows % 16 == 0, 8 waves/block, no divergence
// before the WMMAs (EXEC must be all ones).
// -------------------------------------------------------------------------
__global__ void pool25_wmma(const float* __restrict__ src,
                            float* __restrict__ dst) {
    const int wave = (blockIdx.x * blockDim.x + threadIdx.x) >> 5;
    const int lane = threadIdx.x & 31;
    const int row  = wave * 16 + (lane & 15);
    const int half = lane >> 4;                 // 0: lanes 0-15, 1: lanes 16-31

    const float* rp = src + (size_t)row * HW;

    v2f a; a.x = 1.0f; a.y = 1.0f;              // A = ones
    v8f c = {};
#pragma unroll
    for (int ch = 0; ch < 6; ++ch) {
        const int kb = ch * 4 + half * 2;       // this lane supplies 2 of the
        v2f b; b.x = rp[kb]; b.y = rp[kb + 1];  // 4 K-values of column lane%16
        c = __builtin_amdgcn_wmma_f32_16x16x4_f32(
                /*neg_a=*/false, a, /*neg_b=*/false, b,
                /*c_mod=*/(short)0, c, /*reuse_a=*/false, /*reuse_b=*/false);
    }
    const float sum = c[0] + rp[24];            // remainder element K=24
    if (half == 0) dst[row] = sum * (1.0f / 25.0f);
}

// -------------------------------------------------------------------------
// Average SHOT shots into per-class prototypes: (t,25,64) -> (t,5,64)
// -------------------------------------------------------------------------
__global__ void proto_avg(const float* __restrict__ sf_raw,
                          float* __restrict__ sfp) {
    const int idx = blockIdx.x * blockDim.x + threadIdx.x;  // 0..1279
    if (idx >= NS * C_DIM) return;
    const int ch = idx & (C_DIM - 1);
    const int r  = idx >> 6;                    // t*WAY + w
    const int t  = r / WAY;
    const int w  = r % WAY;
    float s = 0.0f;
#pragma unroll
    for (int sh = 0; sh < SHOT; ++sh)
        s += sf_raw[((t * WS) + (w * SHOT + sh)) * C_DIM + ch];
    sfp[idx] = s * (1.0f / SHOT);
}

// -------------------------------------------------------------------------
// Build (i,j) pair table and support pair-diffs sd[p][s] (p-major).
// -------------------------------------------------------------------------
__global__ void make_pairs_sd(const float* __restrict__ sfp,
                              float* __restrict__ sd,
                              int* __restrict__ pij) {
    const int p = blockIdx.x * blockDim.x + threadIdx.x;
    if (p >= NPAIR) return;
    int i = 0, rem = p;
    while (rem >= (C_DIM - 1) - i) { rem -= (C_DIM - 1) - i; ++i; }
    const int j = i + 1 + rem;
    pij[2 * p]     = i;
    pij[2 * p + 1] = j;
#pragma unroll
    for (int s = 0; s < NS; ++s)
        sd[p * NS + s] = sfp[s * C_DIM + j] - sfp[s * C_DIM + i];
}

// -------------------------------------------------------------------------
// Main: score[q][s] = (1/T) * mean_p (2*sigmoid(qd_p * sd_p) - 1)
// One block per query row. 2*sigmoid(x)-1 = (1-e)*rcp(1+e), e = exp2(-x*log2e)
// -------------------------------------------------------------------------
__global__ void score_kernel(const float* __restrict__ qf,
                             const float* __restrict__ sd,
                             const int* __restrict__ pij,
                             float* __restrict__ out) {
    __shared__ float lqf[C_DIM];
    __shared__ float lacc[NS];
    const int q = blockIdx.x;
    const int t = threadIdx.x;

    if (t < C_DIM) lqf[t] = qf[q * C_DIM + t];
    if (t < NS)    lacc[t] = 0.0f;
    __syncthreads();

    float acc[NS];
#pragma unroll
    for (int s = 0; s < NS; ++s) acc[s] = 0.0f;

    for (int p = t; p < NPAIR; p += 256) {
        const int i = pij[2 * p];
        const int j = pij[2 * p + 1];
        const float qd = lqf[j] - lqf[i];
#pragma unroll
        for (int s = 0; s < NS; ++s) {
            float x = qd * sd[p * NS + s];
            x = fminf(30.0f, fmaxf(-30.0f, x));
            const float e = __builtin_amdgcn_exp2f(x * (-LOG2E));
            acc[s] += (1.0f - e) * __builtin_amdgcn_rcpf(1.0f + e);
        }
    }

#pragma unroll
    for (int s = 0; s < NS; ++s) atomicAdd(&lacc[s], acc[s]);  // ds_add_f32
    __syncthreads();

    if (t < NS)
        out[q * NS + t] = lacc[t] * (80.0f / (float)NPAIR);   // (1/P)/T, T=0.0125
}

// -------------------------------------------------------------------------
extern "C" void kernel_launch(void* const* d_in, const int* in_sizes, int n_in,
                              void* d_out, int out_size, void* d_ws, size_t ws_size,
                              hipStream_t stream) {
    const float* query   = (const float*)d_in[0];   // (4,375,64,5,5)
    const float* support = (const float*)d_in[1];   // (4,25,64,5,5)

    float* ws     = (float*)d_ws;
    float* qf     = ws;                    // 96000 floats  (1500 x 64)
    float* sf_raw = ws + 96000;            //  6400 floats  (100 x 64)
    float* sfp    = ws + 102400;           //  1280 floats  (20 x 64)
    float* sd     = ws + 103680;           // 40320 floats  (2016 x 20)
    int*   pij    = (int*)(ws + 144000);   //  4032 ints
    // total: ~592 KB of workspace

    // Pooling: 96000 rows -> 6000 waves -> 750 blocks; 6400 rows -> 50 blocks
    pool25_wmma<<<750, 256, 0, stream>>>(query, qf);
    pool25_wmma<<<50, 256, 0, stream>>>(support, sf_raw);
    proto_avg<<<5, 256, 0, stream>>>(sf_raw, sfp);
    make_pairs_sd<<<(NPAIR + 255) / 256, 256, 0, stream>>>(sfp, sd, pij);
    score_kernel<<<NQ, 256, 0, stream>>>(qf, sd, pij, (float*)d_out);
}